// Semantic_Mapping_64415919506005
// MI455X (gfx1250) — compile-verified
//
#include <hip/hip_runtime.h>
#include <math.h>

// ---------------- problem constants (match reference) ----------------
#define Bn    4
#define Hh    480
#define Ww    640
#define NSEMc 16
#define Cch   20          // 4 + NSEM
#define Mm    480         // map size
#define VRr   100         // vision range cells
#define NZz   80          // z bins
#define FCH   17          // 1 + NSEM splat feature channels
#define MINZI 13          // agent-height band [13,25)
#define MAXZI 25
#define RAD2DEGf 57.29577951308232f
#define DEG2RADf 0.017453292519943295f

// output layout (floats): fp_map_pred | map_pred | current_poses
#define FP_N   (Bn * VRr * VRr)                 // 40,000
#define MAP_N  ((size_t)Bn * Cch * Mm * Mm)     // 18,432,000

// workspace layout (floats)
#define VOX_N  ((size_t)Bn * FCH * VRr * VRr * NZz)  // 54,400,000
#define AV_N   ((size_t)Bn * Cch * Mm * Mm)          // 18,432,000

#define TPAD  84   // LDS tile row stride in floats (336B: 16B aligned, bank-conflict free)

typedef float v2f __attribute__((ext_vector_type(2)));
typedef float v8f __attribute__((ext_vector_type(8)));

// ---------------------------------------------------------------------
// 1) Point cloud + trilinear splat (scatter-add into voxel grid).
//    vox layout: [b][f][d0=x][d1=depth][z]
// ---------------------------------------------------------------------
__global__ void splat_kernel(const float* __restrict__ obs,
                             float* __restrict__ vox) {
    int tid = blockIdx.x * blockDim.x + threadIdx.x;
    if (tid >= Bn * Hh * Ww) return;
    int w = tid % Ww;
    int h = (tid / Ww) % Hh;
    int b = tid / (Ww * Hh);

    const float focal = 320.0f / tanf(39.5f * DEG2RADf);
    float depth = obs[((size_t)(b * Cch + 3) * Hh + h) * Ww + w];

    // camera frame: X lateral, Y=depth, Z up (row h maps to gz = H-1-h)
    float X  = ((float)w - 319.5f) * depth / focal;
    float Zc = ((float)(Hh - 1 - h) - 239.5f) * depth / focal;

    // splat positions in voxel units (algebraically simplified from reference)
    float pos0 = (X + 250.0f) * 0.2f;        // x / RES shifted
    float pos1 = depth * 0.2f;               // depth / RES
    float pos2 = (Zc + 88.0f) * 0.2f + 8.0f; // z bins, MIN_H offset

    float f0 = floorf(pos0), f1 = floorf(pos1), f2 = floorf(pos2);

    float sem[NSEMc];
#pragma unroll
    for (int s = 0; s < NSEMc; ++s)
        sem[s] = obs[((size_t)(b * Cch + 4 + s) * Hh + h) * Ww + w];

    float* voxb = vox + (size_t)b * FCH * VRr * VRr * NZz;

#pragma unroll
    for (int o = 0; o < 8; ++o) {
        float p0 = f0 + (float)((o >> 2) & 1);
        float p1 = f1 + (float)((o >> 1) & 1);
        float p2 = f2 + (float)(o & 1);
        // reference: safe = (p > 0) & (p < dim)  (strict on both sides)
        if (p0 > 0.f && p0 < (float)VRr &&
            p1 > 0.f && p1 < (float)VRr &&
            p2 > 0.f && p2 < (float)NZz) {
            float wt = (1.f - fabsf(pos0 - p0)) *
                       (1.f - fabsf(pos1 - p1)) *
                       (1.f - fabsf(pos2 - p2));
            size_t cell = (((size_t)p0) * VRr + (size_t)p1) * NZz + (size_t)p2;
            atomicAdd(voxb + cell, wt); // feature channel 0 == ones
#pragma unroll
            for (int s = 0; s < NSEMc; ++s)
                atomicAdd(voxb + (size_t)(s + 1) * VRr * VRr * NZz + cell,
                          sem[s] * wt);
        }
    }
}

// ---------------------------------------------------------------------
// 2) WMMA z-projection with async-to-LDS staging.
//    Per (b,f,row r=d1): stage a 16x80 f32 tile into LDS with
//    GLOBAL_LOAD_ASYNC_TO_LDS_B128 (512B per instruction, split across tile rows,
//    ASYNCcnt tracked), then feed A fragments from LDS into
//    v_wmma_f32_16x16x4_f32. B (4x16): col0=ones (all-height),
//    col1 = indicator z in [13,25) (agent-height). 20 accumulating
//    WMMAs cover K=80.
// ---------------------------------------------------------------------
__global__ void project_wmma_kernel(const float* __restrict__ vox,
                                    float* __restrict__ agent_view,
                                    float* __restrict__ fp_out) {
    __shared__ float tile[16 * TPAD];   // 5376 B staging tile
    __shared__ float dbuf[32 * 8];      // D-matrix spill for extraction

    int lane = threadIdx.x;          // block = 32 = one wave
    int gidx = blockIdx.x;
    int group = gidx % 7;  gidx /= 7;    // 7 groups of 16 cover 100 cols
    int r     = gidx % VRr; gidx /= VRr; // d1 index (output row)
    int f     = gidx % FCH;
    int b     = gidx / FCH;

    int m    = lane & 15;   // A row (pixel in group) / B column
    int half = lane >> 4;   // K sub-pair selector per ISA A layout

    const float* plane = vox + (((size_t)(b * FCH + f)) * VRr) * VRr * NZz;

    // ---- stage 16 rows x 320B into LDS: 320 slots of 16B, 10 waves ----
#if defined(__gfx1250__)
    {
        unsigned lds0 = (unsigned)(size_t)(&tile[0]);
#pragma unroll
        for (int i = 0; i < 10; ++i) {
            int s   = i * 32 + lane;
            int row = s / 20;            // 20 x 16B slots per 320B row
            int zo  = (s % 20) * 4;      // float offset within row
            int c   = group * 16 + row;
            if (c > VRr - 1) c = VRr - 1;    // clamp tail; results discarded
            const float* gsrc = plane + ((size_t)c * VRr + r) * NZz + zo;
            unsigned ldst = lds0 + (unsigned)(row * TPAD + zo) * 4u;
            asm volatile("global_load_async_to_lds_b128 %0, %1, off"
                         :: "v"(ldst),
                            "v"((unsigned long long)(size_t)gsrc)
                         : "memory");
        }
        asm volatile("s_wait_asynccnt 0x0" ::: "memory");
    }
#else
#pragma unroll
    for (int i = 0; i < 10; ++i) {
        int s   = i * 32 + lane;
        int row = s / 20;
        int zo  = (s % 20) * 4;
        int c   = group * 16 + row;
        if (c > VRr - 1) c = VRr - 1;
        const float* gsrc = plane + ((size_t)c * VRr + r) * NZz + zo;
        *reinterpret_cast<float4*>(&tile[row * TPAD + zo]) =
            *reinterpret_cast<const float4*>(gsrc);
    }
#endif
    __syncthreads();

    v8f acc = {};
#pragma unroll
    for (int t = 0; t < 20; ++t) {
        int k0 = t * 4 + half * 2;           // this lane's K pair
        float2 ld = *reinterpret_cast<const float2*>(&tile[m * TPAD + k0]);
        v2f a;
        a.x = rintf(ld.x);                   // jnp.round == RNE
        a.y = rintf(ld.y);
        v2f bm;
        int k1 = k0 + 1;
        bm.x = (m == 0) ? 1.f : ((m == 1 && k0 >= MINZI && k0 < MAXZI) ? 1.f : 0.f);
        bm.y = (m == 0) ? 1.f : ((m == 1 && k1 >= MINZI && k1 < MAXZI) ? 1.f : 0.f);
        // D = A*B + C  (8 args: neg_a, A, neg_b, B, c_mod, C, reuse_a, reuse_b)
        acc = __builtin_amdgcn_wmma_f32_16x16x4_f32(
            false, a, false, bm, (short)0, acc, false, false);
    }

#pragma unroll
    for (int j = 0; j < 8; ++j) dbuf[lane * 8 + j] = acc[j];
    __syncthreads();

    if (lane < 16) {
        int c = group * 16 + lane;
        if (c < VRr) {
            // D layout: vgpr j holds M=j (lanes0-15) / M=8+j (lanes16-31)
            int jm    = lane & 7;
            int baseL = (lane < 8) ? 0 : 16;
            float fullS = dbuf[baseL * 8 + jm];       // N=0: all-height sum
            float rngS  = dbuf[(baseL + 1) * 8 + jm]; // N=1: agent-height sum

            int row = 240 + r;   // y1 = M/2
            int col = 190 + c;   // x1 = M/2 - VR/2
            size_t av0 = ((size_t)b * Cch) * Mm * Mm + (size_t)row * Mm + col;
            if (f == 0) {
                float mp = fminf(fmaxf(rngS, 0.f), 1.f);  // /MAP_THR=1
                float ep = fminf(fmaxf(fullS, 0.f), 1.f); // /EXP_THR=1
                fp_out[(size_t)b * VRr * VRr + (size_t)r * VRr + c] = mp;
                agent_view[av0] = mp;                                 // ch 0
                agent_view[av0 + (size_t)1 * Mm * Mm] = ep;           // ch 1
            } else {
                float cp = fminf(fmaxf(rngS * 0.2f, 0.f), 1.f);       // /CAT_THR=5
                agent_view[av0 + (size_t)(f + 3) * Mm * Mm] = cp;     // ch 4..19
            }
        }
    }
}

// ---------------------------------------------------------------------
// 3) Pose composition + affine parameters
// ---------------------------------------------------------------------
__global__ void pose_kernel(const float* __restrict__ pose_obs,
                            const float* __restrict__ poses_last,
                            float* __restrict__ poses_out,
                            float* __restrict__ theta) {
    int b = threadIdx.x;
    if (b >= Bn) return;
    float plx = poses_last[b * 3 + 0];
    float ply = poses_last[b * 3 + 1];
    float plo = poses_last[b * 3 + 2];
    float p0 = pose_obs[b * 3 + 0];
    float p1 = pose_obs[b * 3 + 1];
    float p2 = pose_obs[b * 3 + 2];

    float rr = plo / RAD2DEGf;
    float sr = sinf(rr), cr = cosf(rr);
    float yy = ply + p0 * sr + p1 * cr;
    float xx = plx + p0 * cr - p1 * sr;
    float oo = plo + p2 * RAD2DEGf;
    oo = fmodf(oo - 180.f, 360.f) + 180.f;
    oo = fmodf(oo + 180.f, 360.f) - 180.f;

    poses_out[b * 3 + 0] = xx;
    poses_out[b * 3 + 1] = yy;
    poses_out[b * 3 + 2] = oo;

    float tx = -(xx * 20.f - 240.f) / 240.f;  // -(x*100/RES - M/2)/(M/2)
    float ty = -(yy * 20.f - 240.f) / 240.f;
    float st = (90.f - oo) * DEG2RADf;
    theta[b * 4 + 0] = cosf(st);
    theta[b * 4 + 1] = sinf(st);
    theta[b * 4 + 2] = tx;
    theta[b * 4 + 3] = ty;
}

// ---------------------------------------------------------------------
// bilinear sampler (align_corners=True, zero padding) on 480x480 plane
// ---------------------------------------------------------------------
__device__ __forceinline__ float fetch480(const float* img, float fx, float fy) {
    bool v = (fx >= 0.f) && (fx <= 479.f) && (fy >= 0.f) && (fy <= 479.f);
    int xi = (int)fminf(fmaxf(fx, 0.f), 479.f);
    int yi = (int)fminf(fmaxf(fy, 0.f), 479.f);
    float val = img[(size_t)yi * Mm + xi];
    return v ? val : 0.f;
}

__device__ __forceinline__ float bilinear480(const float* img, float sx, float sy) {
    float x = (sx + 1.f) * 0.5f * 479.f;
    float y = (sy + 1.f) * 0.5f * 479.f;
    float x0 = floorf(x), y0 = floorf(y);
    float x1 = x0 + 1.f, y1 = y0 + 1.f;
    float wa = (x1 - x) * (y1 - y);
    float wb = (x1 - x) * (y - y0);
    float wc = (x - x0) * (y1 - y);
    float wd = (x - x0) * (y - y0);
    return fetch480(img, x0, y0) * wa + fetch480(img, x0, y1) * wb +
           fetch480(img, x1, y0) * wc + fetch480(img, x1, y1) * wd;
}

// ---------------------------------------------------------------------
// 4) rotation resample
// ---------------------------------------------------------------------
__global__ void rotate_kernel(const float* __restrict__ agent_view,
                              const float* __restrict__ theta,
                              float* __restrict__ rotated) {
    size_t tid = (size_t)blockIdx.x * blockDim.x + threadIdx.x;
    if (tid >= MAP_N) return;
    int x  = (int)(tid % Mm);
    int y  = (int)((tid / Mm) % Mm);
    int ch = (int)((tid / ((size_t)Mm * Mm)) % Cch);
    int b  = (int)(tid / ((size_t)Mm * Mm * Cch));

    float gx = -1.f + 2.f * (float)x / 479.f;
    float gy = -1.f + 2.f * (float)y / 479.f;
    float ct = theta[b * 4 + 0], st = theta[b * 4 + 1];
    float sx = ct * gx - st * gy;
    float sy = st * gx + ct * gy;

    const float* img = agent_view + ((size_t)b * Cch + ch) * Mm * Mm;
    rotated[tid] = bilinear480(img, sx, sy);
}

// ---------------------------------------------------------------------
// 5) translation resample fused with max(maps_last, .)
// ---------------------------------------------------------------------
__global__ void translate_max_kernel(const float* __restrict__ rotated,
                                     const float* __restrict__ maps_last,
                                     const float* __restrict__ theta,
                                     float* __restrict__ map_out) {
    size_t tid = (size_t)blockIdx.x * blockDim.x + threadIdx.x;
    if (tid >= MAP_N) return;
    int x  = (int)(tid % Mm);
    int y  = (int)((tid / Mm) % Mm);
    int ch = (int)((tid / ((size_t)Mm * Mm)) % Cch);
    int b  = (int)(tid / ((size_t)Mm * Mm * Cch));

    float gx = -1.f + 2.f * (float)x / 479.f;
    float gy = -1.f + 2.f * (float)y / 479.f;
    float sx = gx + theta[b * 4 + 2];
    float sy = gy + theta[b * 4 + 3];

    const float* img = rotated + ((size_t)b * Cch + ch) * Mm * Mm;
    float v = bilinear480(img, sx, sy);
    map_out[tid] = fmaxf(maps_last[tid], v);
}

// ---------------------------------------------------------------------
extern "C" void kernel_launch(void* const* d_in, const int* in_sizes, int n_in,
                              void* d_out, int out_size, void* d_ws, size_t ws_size,
                              hipStream_t stream) {
    const float* obs        = (const float*)d_in[0];
    const float* pose_obs   = (const float*)d_in[1];
    const float* maps_last  = (const float*)d_in[2];
    const float* poses_last = (const float*)d_in[3];
    float* out = (float*)d_out;
    float* ws  = (float*)d_ws;

    float* vox   = ws;                 // 54,400,000 f
    float* av    = vox + VOX_N;        // 18,432,000 f
    float* rot   = av + AV_N;          // 18,432,000 f
    float* theta = rot + AV_N;         // 16 f

    // voxel grid + agent_view must start at zero every launch
    hipMemsetAsync(vox, 0, (VOX_N + AV_N) * sizeof(float), stream);

    splat_kernel<<<(Bn * Hh * Ww + 255) / 256, 256, 0, stream>>>(obs, vox);

    project_wmma_kernel<<<Bn * FCH * VRr * 7, 32, 0, stream>>>(vox, av, out);

    pose_kernel<<<1, 32, 0, stream>>>(pose_obs, poses_last,
                                      out + FP_N + MAP_N, theta);

    rotate_kernel<<<(unsigned)((MAP_N + 255) / 256), 256, 0, stream>>>(av, theta, rot);

    translate_max_kernel<<<(unsigned)((MAP_N + 255) / 256), 256, 0, stream>>>(
        rot, maps_last, theta, out + FP_N);
}